// CartBondedWholePoseScoringModule_46643344835027
// MI455X (gfx1250) — compile-verified
//
#include <hip/hip_runtime.h>
#include <cstdint>
#include <cstddef>

#define BLK 256
#define MAXP 64   // max poses supported (reference uses 16)

struct F3 { float x, y, z; };
__device__ __forceinline__ F3 sub3(F3 a, F3 b) { return {a.x - b.x, a.y - b.y, a.z - b.z}; }
__device__ __forceinline__ float dot3(F3 a, F3 b) { return a.x * b.x + a.y * b.y + a.z * b.z; }
__device__ __forceinline__ F3 cross3(F3 a, F3 b) {
    return {a.y * b.z - a.z * b.y, a.z * b.x - a.x * b.z, a.x * b.y - a.y * b.x};
}

__global__ void __launch_bounds__(BLK)
cart_bonded_kernel(const float* __restrict__ coords,
                   const int*   __restrict__ uids,
                   const int*   __restrict__ hkeys,
                   const float* __restrict__ hvals,
                   const int*   __restrict__ subs,   // nsub x 4 int32
                   const int*   __restrict__ offs,   // nposes+1 int32 (uniform -> SGPR loads)
                   double*      __restrict__ acc_out,// nposes doubles, pre-zeroed
                   int nsub, int nposes, int natoms, int nslots)
{
    __shared__ int4  s_sub[BLK];
    __shared__ float s_acc[MAXP];

    const int tid  = threadIdx.x;
    const int i    = blockIdx.x * BLK + tid;
    const int gidx = (i < nsub) ? i : (nsub - 1);   // clamp tail lanes (data ignored later)

    // --- CDNA5 async global->LDS staging of this block's subgraph rows (16B/lane) ---
    {
        unsigned lds_addr = (unsigned)(size_t)(&s_sub[tid]);   // low 32 bits = LDS byte offset
        unsigned voff     = (unsigned)gidx * 16u;              // byte offset into subs
        asm volatile("global_load_async_to_lds_b128 %0, %1, %2"
                     :: "v"(lds_addr), "v"(voff), "s"(subs) : "memory");
    }

    if (tid < MAXP) s_acc[tid] = 0.0f;

    // pid = clip(searchsorted(offs, i, 'right') - 1, 0, nposes-1)
    // offs[j] is wave-uniform -> scalar loads + per-lane compare/accumulate.
    int pid = 0;
    #pragma unroll 4
    for (int j = 1; j <= nposes; ++j)
        pid += (offs[j] <= i) ? 1 : 0;
    if (pid > nposes - 1) pid = nposes - 1;

    asm volatile("s_wait_asynccnt 0" ::: "memory");
    __syncthreads();

    if (i < nsub) {
        const int4 q = s_sub[tid];

        const bool v0 = q.x >= 0, v1 = q.y >= 0, v2 = q.z >= 0, v3 = q.w >= 0;
        const int nat = (int)v0 + (int)v1 + (int)v2 + (int)v3;
        const int i0 = v0 ? q.x : 0, i1 = v1 ? q.y : 0, i2 = v2 ? q.z : 0, i3 = v3 ? q.w : 0;

        const size_t pbase = (size_t)pid * (size_t)natoms;
        const float* cb = coords + pbase * 3;
        auto ldc = [&](int a) -> F3 {
            const float* p = cb + (size_t)a * 3;
            return {p[0], p[1], p[2]};
        };

        // unique-id gathers (zeros for invalid atoms, matching reference)
        const int* ub = uids + pbase;
        const int u0 = v0 ? ub[i0] : 0;
        const int u1 = v1 ? ub[i1] : 0;
        const int u2 = v2 ? ub[i2] : 0;
        const int u3 = v3 ? ub[i3] : 0;

        const int key = (u0 * 3 + u1 * 5 + u2 * 7 + u3 * 11 + nat) % nslots;
        const bool hit = (hkeys[key] == key);
        const float Kc = hit ? hvals[3 * key + 0] : 1.0f;
        const float x0 = hit ? hvals[3 * key + 1] : 0.0f;

        const float eps = 1e-6f;
        float e;

        // nat is contiguous over the subgraph index range (thirds), so these
        // branches are wave-uniform except at two boundary waves in the grid.
        if (nat == 2) {
            // bond: only p0,p1 needed (skip half the coord gather traffic)
            const F3 p0 = ldc(i0), p1 = ldc(i1);
            const F3 d10 = sub3(p1, p0);
            const float d = sqrtf(dot3(d10, d10) + eps);
            e = Kc * (d - x0) * (d - x0);
        } else if (nat == 3) {
            // angle: p0,p1,p2
            const F3 p0 = ldc(i0), p1 = ldc(i1), p2 = ldc(i2);
            const F3 uu = sub3(p0, p1), vv = sub3(p2, p1);
            float ca = dot3(uu, vv) /
                       (sqrtf(dot3(uu, uu) + eps) * sqrtf(dot3(vv, vv) + eps));
            ca = fminf(fmaxf(ca, -1.0f + 1e-6f), 1.0f - 1e-6f);
            const float th = acosf(ca);
            e = Kc * (th - x0) * (th - x0);
        } else {
            // torsion: all four atoms
            const F3 p0 = ldc(i0), p1 = ldc(i1), p2 = ldc(i2), p3 = ldc(i3);
            const float per = hit ? (rintf(hvals[3 * key + 2] * 3.0f) + 1.0f) : 1.0f;
            const F3 b1 = sub3(p1, p0), b2 = sub3(p2, p1), b3 = sub3(p3, p2);
            const F3 n1 = cross3(b1, b2), n2 = cross3(b2, b3);
            const float sb = sqrtf(dot3(b2, b2) + eps);
            const F3 b2u = {b2.x / sb, b2.y / sb, b2.z / sb};
            const F3 m1 = cross3(n1, b2u);
            const float phi = atan2f(dot3(m1, n2), dot3(n1, n2) + eps);
            e = Kc * (1.0f + cosf(per * phi - x0));
        }

        atomicAdd(&s_acc[pid], e);   // ds_add_f32
    }

    __syncthreads();

    if (tid < nposes) {
        const float p = s_acc[tid];
        if (p != 0.0f)
            atomicAdd(&acc_out[tid], (double)p);  // global_atomic_add_f64, 16 slots
    }
}

__global__ void finalize_kernel(const double* __restrict__ ws, float* __restrict__ out, int nposes)
{
    const int t = blockIdx.x * blockDim.x + threadIdx.x;
    if (t < nposes) out[t] = (float)ws[t];
}

extern "C" void kernel_launch(void* const* d_in, const int* in_sizes, int n_in,
                              void* d_out, int out_size, void* d_ws, size_t ws_size,
                              hipStream_t stream)
{
    // setup_inputs order:
    // 0 coords(f32), 1 block_coord_offset, 2 block_types, 3 inter_block_conn,
    // 4 atom_paths_from_conn, 5 atom_unique_ids(i32), 6 hash_keys(i32),
    // 7 hash_values(f32), 8 cart_subgraphs(i32), 9 cart_subgraph_offsets(i32)
    const float* coords = (const float*)d_in[0];
    const int*   uids   = (const int*)d_in[5];
    const int*   hkeys  = (const int*)d_in[6];
    const float* hvals  = (const float*)d_in[7];
    const int*   subs   = (const int*)d_in[8];
    const int*   offs   = (const int*)d_in[9];

    const int nslots = in_sizes[6];
    const int nsub   = in_sizes[8] / 4;
    const int nposes = in_sizes[9] - 1;
    const int natoms = in_sizes[5] / nposes;

    double* ws = (double*)d_ws;
    hipMemsetAsync(ws, 0, (size_t)nposes * sizeof(double), stream);

    const int grid = (nsub + BLK - 1) / BLK;
    hipLaunchKernelGGL(cart_bonded_kernel, dim3(grid), dim3(BLK), 0, stream,
                       coords, uids, hkeys, hvals, subs, offs, ws,
                       nsub, nposes, natoms, nslots);

    hipLaunchKernelGGL(finalize_kernel, dim3(1), dim3(MAXP), 0, stream,
                       ws, (float*)d_out, nposes);
}